// MultiHeadedAttention_41669772705900
// MI455X (gfx1250) — compile-verified
//
#include <hip/hip_runtime.h>
#include <hip/hip_bf16.h>

typedef __attribute__((ext_vector_type(2))) float v2f;
typedef __attribute__((ext_vector_type(8))) float v8f;

#define HID 768
#define SEQ 2048
#define NB  2
#define NH  12
#define HD  64
#define QKV_ELEMS ((size_t)NB * SEQ * HID)   // 3,145,728 floats per matrix

// ---------------------------------------------------------------------------
// Kernel 1: Y = X @ W + b for W in {Wq,Wk,Wv} (blockIdx.z selects), writing
// result directly in [B, H, S, D] layout into workspace.
// Block = 256 threads = 8 waves; wave w computes rows [m0+16w, m0+16w+16) of a
// 128-row x 64-col output tile.  W K-chunk (32x64) staged in LDS transposed.
// ---------------------------------------------------------------------------
__global__ __launch_bounds__(256) void qkv_proj_kernel(
    const float* __restrict__ X,
    const float* __restrict__ Wq, const float* __restrict__ bq,
    const float* __restrict__ Wk, const float* __restrict__ bk,
    const float* __restrict__ Wv, const float* __restrict__ bv,
    float* __restrict__ ws)
{
    __shared__ float lw[64 * 34];            // transposed W tile, padded stride
    const int tid  = threadIdx.x;
    const int wave = tid >> 5;
    const int lane = tid & 31;
    const int l15  = lane & 15;
    const int hw   = lane >> 4;              // half-wave: 0 or 1

    const int zsel = blockIdx.z;
    const float* W    = (zsel == 0) ? Wq : (zsel == 1) ? Wk : Wv;
    const float* bias = (zsel == 0) ? bq : (zsel == 1) ? bk : bv;
    float* Y = ws + (size_t)zsel * QKV_ELEMS;

    const int head = blockIdx.x;             // 0..11, n0 = head*64
    const int n0   = head * 64;
    const int m0   = blockIdx.y * 128 + wave * 16;

    v8f acc[4];
    #pragma unroll
    for (int t = 0; t < 4; ++t) { v8f zz = {}; acc[t] = zz; }

    for (int kc0 = 0; kc0 < HID; kc0 += 32) {
        __syncthreads();
        // cooperative load of W[kc0:kc0+32, n0:n0+64] transposed into LDS
        #pragma unroll
        for (int i = 0; i < 8; ++i) {
            int lin = tid + 256 * i;         // 0..2047 covers 32x64
            int kr  = lin >> 6;              // 0..31
            int nc  = lin & 63;              // 0..63
            lw[nc * 34 + kr] = W[(size_t)(kc0 + kr) * HID + n0 + nc];
        }
        __syncthreads();

        #pragma unroll
        for (int kc = 0; kc < 8; ++kc) {
            const int kk = kc * 4 + 2 * hw;  // even
            // A fragment: X[m0+l15, kc0+kk .. +1]  (contiguous float2)
            v2f a = *(const v2f*)(X + (size_t)(m0 + l15) * HID + kc0 + kk);
            #pragma unroll
            for (int t = 0; t < 4; ++t) {
                // B fragment from transposed LDS: contiguous b64
                v2f b = *(const v2f*)(lw + (16 * t + l15) * 34 + kk);
                acc[t] = __builtin_amdgcn_wmma_f32_16x16x4_f32(
                    false, a, false, b, (short)0, acc[t], false, false);
            }
        }
    }

    // epilogue: add bias, store in [B, H, S, D] layout
    #pragma unroll
    for (int t = 0; t < 4; ++t) {
        const int d  = 16 * t + l15;         // 0..63 within head
        const float bb = bias[n0 + d];
        #pragma unroll
        for (int r = 0; r < 8; ++r) {
            const int m   = r + 8 * hw;
            const int row = m0 + m;          // 0..4095, never crosses batch
            const int bat = row >> 11;
            const int s   = row & (SEQ - 1);
            Y[(((size_t)bat * NH + head) * SEQ + s) * HD + d] = acc[t][r] + bb;
        }
    }
}

// ---------------------------------------------------------------------------
// Kernel 2: flash-style attention.  One wave per 16-query tile of one (b,h).
// Q fragments held in registers (pre-scaled by 1/sqrt(64)); online softmax;
// P tile bounced through per-wave padded LDS to convert C-layout -> A-layout.
// ---------------------------------------------------------------------------
__global__ __launch_bounds__(256) void attn_kernel(
    const float* __restrict__ ws,
    const float* __restrict__ mask,          // [B,1,1,S] = B*S floats
    float* __restrict__ out)                 // [B, S, H*D]
{
    __shared__ float lp[8 * 16 * 18];        // per-wave 16x16 P tile, stride 18
    const int tid  = threadIdx.x;
    const int wave = tid >> 5;
    const int lane = tid & 31;
    const int l15  = lane & 15;
    const int hw   = lane >> 4;
    float* pw = lp + wave * (16 * 18);

    const int gw = blockIdx.x * 8 + wave;    // 0..3071
    const int bh = gw >> 7;                  // 0..23
    const int qt = gw & 127;
    const int b  = bh / NH;
    const int h  = bh % NH;
    const int m0 = qt * 16;

    const float* Q = ws + (size_t)bh * SEQ * HD;
    const float* K = ws + QKV_ELEMS     + (size_t)bh * SEQ * HD;
    const float* V = ws + 2 * QKV_ELEMS + (size_t)bh * SEQ * HD;

    // preload Q fragments, scaled by 1/sqrt(HD) = 0.125
    v2f qf[16];
    #pragma unroll
    for (int dc = 0; dc < 16; ++dc) {
        v2f a = *(const v2f*)(Q + (size_t)(m0 + l15) * HD + dc * 4 + 2 * hw);
        qf[dc] = a * 0.125f;
    }

    v8f oacc[4];
    #pragma unroll
    for (int t = 0; t < 4; ++t) { v8f zz = {}; oacc[t] = zz; }
    float mrow[8], lrow[8];
    #pragma unroll
    for (int r = 0; r < 8; ++r) { mrow[r] = -1e30f; lrow[r] = 0.0f; }

    for (int k0 = 0; k0 < SEQ; k0 += 16) {
        // ---- S tile = (Q*0.125) @ K^T  (16x16) ----
        v8f s = {};
        #pragma unroll
        for (int dc = 0; dc < 16; ++dc) {
            // B frag of K^T: K[k0+l15, dc*4 + 2*hw .. +1]  (contiguous float2)
            v2f kb = *(const v2f*)(K + (size_t)(k0 + l15) * HD + dc * 4 + 2 * hw);
            s = __builtin_amdgcn_wmma_f32_16x16x4_f32(
                false, qf[dc], false, kb, (short)0, s, false, false);
        }
        const float mval = mask[b * SEQ + k0 + l15];  // per-key mask value

        // ---- online softmax (row m lives in vgpr m&7, half-wave m>=8) ----
        float oscale[8];
        #pragma unroll
        for (int r = 0; r < 8; ++r) {
            float sv = s[r] + mval;
            float rmax = sv;
            rmax = fmaxf(rmax, __shfl_xor(rmax, 1));
            rmax = fmaxf(rmax, __shfl_xor(rmax, 2));
            rmax = fmaxf(rmax, __shfl_xor(rmax, 4));
            rmax = fmaxf(rmax, __shfl_xor(rmax, 8));
            float mnew = fmaxf(mrow[r], rmax);
            float sc   = __expf(mrow[r] - mnew);
            float p    = __expf(sv - mnew);
            float rs = p;
            rs += __shfl_xor(rs, 1);
            rs += __shfl_xor(rs, 2);
            rs += __shfl_xor(rs, 4);
            rs += __shfl_xor(rs, 8);
            lrow[r] = lrow[r] * sc + rs;
            mrow[r] = mnew;
            oscale[r] = sc;
            // stash P in C-layout into per-wave LDS tile
            pw[(r + 8 * hw) * 18 + l15] = p;
        }

        // rescale running O
        #pragma unroll
        for (int t = 0; t < 4; ++t)
            #pragma unroll
            for (int r = 0; r < 8; ++r)
                oacc[t][r] *= oscale[r];

        // ---- O += P @ V  (16x16 x 16x64) ----
        #pragma unroll
        for (int kc = 0; kc < 4; ++kc) {
            // A frag of P from LDS: row l15, cols kc*4+2*hw .. +1 (b64, even)
            v2f pa = *(const v2f*)(pw + l15 * 18 + kc * 4 + 2 * hw);
            const int krow = k0 + kc * 4 + 2 * hw;
            #pragma unroll
            for (int t = 0; t < 4; ++t) {
                v2f vb;
                vb[0] = V[(size_t)krow * HD + 16 * t + l15];
                vb[1] = V[(size_t)(krow + 1) * HD + 16 * t + l15];
                oacc[t] = __builtin_amdgcn_wmma_f32_16x16x4_f32(
                    false, pa, false, vb, (short)0, oacc[t], false, false);
            }
        }
    }

    // ---- finalize: O /= l, scatter to [B, S, H*D] ----
    #pragma unroll
    for (int t = 0; t < 4; ++t) {
        const int d = 16 * t + l15;
        #pragma unroll
        for (int r = 0; r < 8; ++r) {
            const int m = r + 8 * hw;
            out[((size_t)(b * SEQ + m0 + m)) * HID + h * HD + d] =
                oacc[t][r] / lrow[r];
        }
    }
}

extern "C" void kernel_launch(void* const* d_in, const int* in_sizes, int n_in,
                              void* d_out, int out_size, void* d_ws, size_t ws_size,
                              hipStream_t stream) {
    const float* hidden = (const float*)d_in[0];
    const float* amask  = (const float*)d_in[1];
    const float* Wq = (const float*)d_in[2];
    const float* bq = (const float*)d_in[3];
    const float* Wk = (const float*)d_in[4];
    const float* bk = (const float*)d_in[5];
    const float* Wv = (const float*)d_in[6];
    const float* bv = (const float*)d_in[7];
    float* out = (float*)d_out;
    float* ws  = (float*)d_ws;    // needs 3 * 4096*768*4 = 37.75 MB

    // QKV projections: N-tiles (12 heads) x M-tiles (4096/128) x {q,k,v}
    qkv_proj_kernel<<<dim3(NH, (NB * SEQ) / 128, 3), 256, 0, stream>>>(
        hidden, Wq, bq, Wk, bk, Wv, bv, ws);

    // Attention: 3072 query tiles, 8 waves per block
    attn_kernel<<<(NB * NH * (SEQ / 16)) / 8, 256, 0, stream>>>(ws, amask, out);
}